// BaselineGraphSAGE_82188494176333
// MI455X (gfx1250) — compile-verified
//
#include <hip/hip_runtime.h>

// ---------------------------------------------------------------------------
// GraphSAGE (3 layers, eval) for MI455X / gfx1250, wave32.
//   per layer: msg[dst] += x[src]  (atomic scatter, f32)
//              h = (msg/deg) @ W_l + b + x @ W_r   [fused WMMA f32 GEMM]
//              relu (layers 0,1)
// Degree computed once (dst is layer-invariant).
// ---------------------------------------------------------------------------

typedef __attribute__((ext_vector_type(2))) float v2f;
typedef __attribute__((ext_vector_type(8))) float v8f;

// ---- degree: deg[dst[e]] += 1 ------------------------------------------------
__global__ void deg_kernel(const long long* __restrict__ dst, float* __restrict__ deg, int nE) {
    int t = blockIdx.x * blockDim.x + threadIdx.x;
    if (t < nE) atomicAdd(&deg[(int)dst[t]], 1.0f);
}

// ---- invdeg[i] = 1 / max(deg[i], 1) -----------------------------------------
__global__ void invdeg_kernel(float* __restrict__ deg, int n) {
    int t = blockIdx.x * blockDim.x + threadIdx.x;
    if (t < n) deg[t] = 1.0f / fmaxf(deg[t], 1.0f);
}

// ---- scatter: one wave per edge, lane L moves features [4L, 4L+3] -----------
// DIN == 128 -> 32 lanes x 4 floats, coalesced float4 gather + 4 f32 atomics.
__global__ __launch_bounds__(256) void sage_scatter128(
        const float* __restrict__ x, const long long* __restrict__ ei,
        float* __restrict__ msg, int nE) {
    int wave = (int)((blockIdx.x * blockDim.x + threadIdx.x) >> 5);
    int lane = threadIdx.x & 31;
    if (wave >= nE) return;
    int s = (int)ei[wave];        // src row
    int d = (int)ei[nE + wave];   // dst row
    const float4 v = *(const float4*)(x + (size_t)s * 128 + lane * 4);
    float* p = msg + (size_t)d * 128 + lane * 4;
    atomicAdd(p + 0, v.x);
    atomicAdd(p + 1, v.y);
    atomicAdd(p + 2, v.z);
    atomicAdd(p + 3, v.w);
}

// ---- fused SAGE GEMM: out = (msg*invdeg) @ Wl + bias + xin @ Wr, opt. ReLU --
// One wave computes one 16x16 output tile with V_WMMA_F32_16X16X4_F32.
// A layout (16x4 f32): lane L -> M = L%16, K pair {0,1} (L<16) / {2,3} (L>=16)
// B layout (4x16 f32): lane L -> N = L%16, same K-pair split
// C/D (16x16 f32, 8 VGPRs): vgpr i -> row i + (L<16?0:8), col L%16
template <int DIN, int DOUT, bool RELU>
__global__ __launch_bounds__(256) void sage_wmma_gemm(
        const float* __restrict__ msg, const float* __restrict__ xin,
        const float* __restrict__ invdeg,
        const float* __restrict__ Wl, const float* __restrict__ bl,
        const float* __restrict__ Wr, float* __restrict__ out) {
    const int lane  = threadIdx.x & 31;
    const int wave  = threadIdx.x >> 5;
    const int mtile = blockIdx.x;                              // rows/16 tiles
    const int ntile = blockIdx.y * (blockDim.x >> 5) + wave;   // cols/16 tiles
    if (ntile * 16 >= DOUT) return;  // wave-uniform: EXEC stays all-ones

    const int hi    = lane >> 4;           // 0 for lanes 0-15, 1 for 16-31
    const int row   = mtile * 16 + (lane & 15);
    const int col   = ntile * 16 + (lane & 15);
    const int khalf = hi * 2;              // K offset {0 | 2} for this lane half

    const float sc = invdeg[row];
    const float* mRow = msg + (size_t)row * DIN + khalf;
    const float* xRow = xin + (size_t)row * DIN + khalf;

    v8f acc = {};
#pragma unroll 4
    for (int k = 0; k < DIN; k += 4) {
        v2f a0, a1, b0, b1;
        a0.x = mRow[k]     * sc;                       // mean tile (fused /deg)
        a0.y = mRow[k + 1] * sc;
        a1.x = xRow[k];                                // self tile
        a1.y = xRow[k + 1];
        b0.x = Wl[(size_t)(k + khalf)     * DOUT + col];
        b0.y = Wl[(size_t)(k + khalf + 1) * DOUT + col];
        b1.x = Wr[(size_t)(k + khalf)     * DOUT + col];
        b1.y = Wr[(size_t)(k + khalf + 1) * DOUT + col];
        // (neg_a, A, neg_b, B, c_mod, C, reuse_a, reuse_b)
        acc = __builtin_amdgcn_wmma_f32_16x16x4_f32(false, a0, false, b0,
                                                    (short)0, acc, false, false);
        acc = __builtin_amdgcn_wmma_f32_16x16x4_f32(false, a1, false, b1,
                                                    (short)0, acc, false, false);
    }

    const float bias  = bl[col];
    const int   rbase = mtile * 16 + (hi << 3);
#pragma unroll
    for (int i = 0; i < 8; ++i) {
        float v = acc[i] + bias;
        if (RELU) v = fmaxf(v, 0.0f);
        out[(size_t)(rbase + i) * DOUT + col] = v;
    }
}

// ---------------------------------------------------------------------------
extern "C" void kernel_launch(void* const* d_in, const int* in_sizes, int n_in,
                              void* d_out, int out_size, void* d_ws, size_t ws_size,
                              hipStream_t stream) {
    const float*     x   = (const float*)d_in[0];
    const long long* ei  = (const long long*)d_in[1];   // int64 [2, E] flat
    const float* Wl0 = (const float*)d_in[2];
    const float* bl0 = (const float*)d_in[3];
    const float* Wr0 = (const float*)d_in[4];
    const float* Wl1 = (const float*)d_in[5];
    const float* bl1 = (const float*)d_in[6];
    const float* Wr1 = (const float*)d_in[7];
    const float* Wl2 = (const float*)d_in[8];
    const float* bl2 = (const float*)d_in[9];
    const float* Wr2 = (const float*)d_in[10];
    float* out = (float*)d_out;

    const int N = in_sizes[0] / 128;   // 50000
    const int E = in_sizes[1] / 2;     // 800000

    // Workspace layout (all f32): invdeg[N] | msg[N*128] | h1[N*128] | h2[N*128]
    char*  ws  = (char*)d_ws;
    size_t off = 0;
    float* deg = (float*)(ws + off); off += ((size_t)N * 4 + 255) & ~(size_t)255;
    float* msg = (float*)(ws + off); off += (size_t)N * 128 * 4;
    float* h1  = (float*)(ws + off); off += (size_t)N * 128 * 4;
    float* h2  = (float*)(ws + off);

    const int mtiles       = N / 16;                  // 3125 (N % 16 == 0)
    const int scatterBlks  = (E + 7) / 8;             // 8 waves (edges) / block
    const int edgeThrBlks  = (E + 255) / 256;
    const int nodeThrBlks  = (N + 255) / 256;

    // Degree (layer-invariant), then reciprocal.
    hipMemsetAsync(deg, 0, (size_t)N * 4, stream);
    deg_kernel<<<edgeThrBlks, 256, 0, stream>>>(ei + E, deg, E);
    invdeg_kernel<<<nodeThrBlks, 256, 0, stream>>>(deg, N);

    // ---- layer 0: 128 -> 128, relu ----
    hipMemsetAsync(msg, 0, (size_t)N * 128 * 4, stream);
    sage_scatter128<<<scatterBlks, 256, 0, stream>>>(x, ei, msg, E);
    sage_wmma_gemm<128, 128, true><<<dim3(mtiles, 1), 256, 0, stream>>>(
        msg, x, deg, Wl0, bl0, Wr0, h1);

    // ---- layer 1: 128 -> 128, relu ----
    hipMemsetAsync(msg, 0, (size_t)N * 128 * 4, stream);
    sage_scatter128<<<scatterBlks, 256, 0, stream>>>(h1, ei, msg, E);
    sage_wmma_gemm<128, 128, true><<<dim3(mtiles, 1), 256, 0, stream>>>(
        msg, h1, deg, Wl1, bl1, Wr1, h2);

    // ---- layer 2: 128 -> 64, no relu ----
    hipMemsetAsync(msg, 0, (size_t)N * 128 * 4, stream);
    sage_scatter128<<<scatterBlks, 256, 0, stream>>>(h2, ei, msg, E);
    sage_wmma_gemm<128, 64, false><<<dim3(mtiles, 1), 128, 0, stream>>>(
        msg, h2, deg, Wl2, bl2, Wr2, out);
}